// MetaOpPolicyNet_45749991637043
// MI455X (gfx1250) — compile-verified
//
#include <hip/hip_runtime.h>
#include <hip/hip_bf16.h>

typedef __attribute__((ext_vector_type(4))) int   v4i;
typedef __attribute__((ext_vector_type(2))) float v2f;
typedef __attribute__((ext_vector_type(8))) float v8f;

// ---------------------------------------------------------------------------
// Kernel 1: per-image color stats (count, ysum, xsum) packed into one u64
// LDS atomic per pixel.  One block = one 128x128 image, 256 threads (8 waves).
//
// Packing:  bits [20:0]  = xsum   (max 16384*127 = 2,080,768 < 2^21)
//           bits [41:21] = ysum   (same bound)
//           bits [63:42] = count  (max 16384 < 2^15)
// No field can carry into the next, so a single ds_add_u64 accumulates all 3.
//
// The 256 MiB grid is read-once against a 192 MB L2 -> load it NON-TEMPORAL
// (TH=NT) so the stream doesn't thrash resident lines.
// ---------------------------------------------------------------------------
__global__ __launch_bounds__(256) void stats_kernel(const int* __restrict__ grid,
                                                    float* __restrict__ stats) {
    // 8 per-wave accumulator regions, 16 u64 slots each (10 used),
    // 128-byte stride so waves land on different bank groups.
    __shared__ unsigned long long acc[8][16];

    const int tid  = threadIdx.x;
    const int wave = tid >> 5;

    if (tid < 128) ((unsigned long long*)acc)[tid] = 0ull;
    __syncthreads();

    const int* __restrict__ g = grid + (size_t)blockIdx.x * 16384;
    unsigned long long* my = acc[wave];

    // 16 iterations x 256 threads x b128 = 16384 pixels, fully coalesced.
    #pragma unroll 4
    for (int i = 0; i < 16; ++i) {
        const int flat0 = i * 1024 + tid * 4;
        const v4i px = __builtin_nontemporal_load((const v4i*)(g + flat0));
        #pragma unroll
        for (int j = 0; j < 4; ++j) {
            const int p = flat0 + j;                 // flat pixel index
            const int c = px[j];                     // color 0..9
            const unsigned long long inc =
                  (unsigned long long)(p & 127)                 // xsum += x
                | ((unsigned long long)(p >> 7) << 21)          // ysum += y
                | (1ull << 42);                                 // count += 1
            atomicAdd(&my[c], inc);                  // ds_add_u64
        }
    }
    __syncthreads();

    if (tid < 10) {
        unsigned long long s = 0;
        #pragma unroll
        for (int w = 0; w < 8; ++w) s += acc[w][tid];
        const float xs  = (float)(s & 0x1FFFFFull);
        const float ys  = (float)((s >> 21) & 0x1FFFFFull);
        const float cnt = (float)(s >> 42);
        const float inv = cnt > 0.0f ? 1.0f / cnt : 0.0f;
        float* o = stats + (size_t)blockIdx.x * 40 + tid * 4;
        o[0] = cnt; o[1] = cnt; o[2] = ys * inv; o[3] = xs * inv;
    }
}

// ---------------------------------------------------------------------------
// Kernel 2: 40->64->32->32 MLP on the stats, one wave (32 threads) per 16
// batch rows, using V_WMMA_F32_16X16X4_F32 (f32 keeps counts exact).
//
// A (16x4) layout per ISA: lanes 0-15 hold row M=lane with K=0,1 in the two
// VGPRs; lanes 16-31 hold K=2,3.  We mirror that for B (4x16).
// C/D (16x16 f32): VGPR j <-> M=j (lanes 0-15) / M=j+8 (lanes 16-31), N=lane&15.
// ---------------------------------------------------------------------------
__global__ __launch_bounds__(32) void mlp_kernel(const float* __restrict__ stats,
                                                 const float* __restrict__ W1,
                                                 const float* __restrict__ b1,
                                                 const float* __restrict__ W2,
                                                 const float* __restrict__ b2,
                                                 const float* __restrict__ W3,
                                                 const float* __restrict__ b3,
                                                 float* __restrict__ out) {
    __shared__ float h1[16 * 64];
    __shared__ float h2[16 * 32];

    const int lane = threadIdx.x;       // 0..31, no divergence anywhere
    const int hv   = lane >> 4;         // 0: K=0,1   1: K=2,3
    const int l15  = lane & 15;
    const int row0 = blockIdx.x * 16;

    // ---------------- Layer 1: (16x40) @ (40x64) + b1, ReLU ----------------
    #pragma unroll
    for (int nt = 0; nt < 4; ++nt) {
        v8f c = {};
        const int n = nt * 16 + l15;
        #pragma unroll
        for (int kc = 0; kc < 10; ++kc) {
            const int k0 = kc * 4 + hv * 2;
            v2f a, b;
            a.x = stats[(row0 + l15) * 40 + k0];
            a.y = stats[(row0 + l15) * 40 + k0 + 1];
            b.x = W1[(k0    ) * 64 + n];
            b.y = W1[(k0 + 1) * 64 + n];
            c = __builtin_amdgcn_wmma_f32_16x16x4_f32(false, a, false, b,
                                                      (short)0, c, false, false);
        }
        const float bias = b1[n];
        #pragma unroll
        for (int j = 0; j < 8; ++j) {
            float v = c[j] + bias;
            h1[(j + hv * 8) * 64 + n] = v > 0.0f ? v : 0.0f;
        }
    }
    __syncthreads();

    // ---------------- Layer 2: (16x64) @ (64x32) + b2, ReLU ----------------
    #pragma unroll
    for (int nt = 0; nt < 2; ++nt) {
        v8f c = {};
        const int n = nt * 16 + l15;
        #pragma unroll
        for (int kc = 0; kc < 16; ++kc) {
            const int k0 = kc * 4 + hv * 2;
            v2f a, b;
            a.x = h1[l15 * 64 + k0];
            a.y = h1[l15 * 64 + k0 + 1];
            b.x = W2[(k0    ) * 32 + n];
            b.y = W2[(k0 + 1) * 32 + n];
            c = __builtin_amdgcn_wmma_f32_16x16x4_f32(false, a, false, b,
                                                      (short)0, c, false, false);
        }
        const float bias = b2[n];
        #pragma unroll
        for (int j = 0; j < 8; ++j) {
            float v = c[j] + bias;
            h2[(j + hv * 8) * 32 + n] = v > 0.0f ? v : 0.0f;
        }
    }
    __syncthreads();

    // ---------------- Layer 3: (16x32) @ (32x32) + b3 ----------------------
    #pragma unroll
    for (int nt = 0; nt < 2; ++nt) {
        v8f c = {};
        const int n = nt * 16 + l15;
        #pragma unroll
        for (int kc = 0; kc < 8; ++kc) {
            const int k0 = kc * 4 + hv * 2;
            v2f a, b;
            a.x = h2[l15 * 32 + k0];
            a.y = h2[l15 * 32 + k0 + 1];
            b.x = W3[(k0    ) * 32 + n];
            b.y = W3[(k0 + 1) * 32 + n];
            c = __builtin_amdgcn_wmma_f32_16x16x4_f32(false, a, false, b,
                                                      (short)0, c, false, false);
        }
        const float bias = b3[n];
        #pragma unroll
        for (int j = 0; j < 8; ++j) {
            out[(size_t)(row0 + j + hv * 8) * 32 + n] = c[j] + bias;
        }
    }
}

// ---------------------------------------------------------------------------
extern "C" void kernel_launch(void* const* d_in, const int* in_sizes, int n_in,
                              void* d_out, int out_size, void* d_ws, size_t ws_size,
                              hipStream_t stream) {
    const int*   grid = (const int*)  d_in[0];
    const float* W1   = (const float*)d_in[1];
    const float* b1   = (const float*)d_in[2];
    const float* W2   = (const float*)d_in[3];
    const float* b2   = (const float*)d_in[4];
    const float* W3   = (const float*)d_in[5];
    const float* b3   = (const float*)d_in[6];
    float*       out  = (float*)d_out;
    float*       stats = (float*)d_ws;          // (B, 40) f32 scratch

    const int B = in_sizes[0] / (128 * 128);    // 4096

    stats_kernel<<<B, 256, 0, stream>>>(grid, stats);
    mlp_kernel<<<B / 16, 32, 0, stream>>>(stats, W1, b1, W2, b2, W3, b3, out);
}